// gious_3D_56453050139230
// MI455X (gfx1250) — compile-verified
//
#include <hip/hip_runtime.h>

// GIoU-3D (aligned boxes, faithful-bug reproduction) for MI455X / gfx1250.
//
// Memory-bound op: 60 B/row, ~90 f32 FLOPs/row -> bandwidth roofline
// (~10 us @ 23.3 TB/s for 4M rows). WMMA is not applicable (elementwise, no
// contraction). CDNA5 feature used: async tensor-data path
// GLOBAL_LOAD_ASYNC_TO_LDS_B128 (+ s_wait_asynccnt) to stage the stride-7
// records into LDS with fully coalesced 16B transfers; LDS row reads at
// stride 7 are bank-conflict-free (gcd(7,64)=1).

#define TPB 256
#define ROWS_PER_BLOCK 256
#define FLOATS_PER_BLOCK (ROWS_PER_BLOCK * 7)   // 1792 floats = 7168 bytes
#define VEC4_PER_BLOCK (FLOATS_PER_BLOCK / 4)   // 448 x 16B chunks

#define GIOU_EPS 1e-16f

__device__ __forceinline__ void async_copy_b128(unsigned lds_byte_off,
                                                const void* gptr) {
  // LDS dest byte-offset in a VGPR, 64-bit global address in a VGPR pair.
  asm volatile("global_load_async_to_lds_b128 %0, %1, off"
               :
               : "v"(lds_byte_off), "v"(gptr)
               : "memory");
}

__device__ __forceinline__ void plane(float gx, float gy, float gdx, float gdy,
                                      float qx, float qy, float qdx, float qdy,
                                      float& inter, float& mbr) {
  float x0g = gx - 0.5f * gdx, x1g = gx + 0.5f * gdx;
  float y0g = gy - 0.5f * gdy, y1g = gy + 0.5f * gdy;
  float x0q = qx - 0.5f * qdx, x1q = qx + 0.5f * qdx;
  float y0q = qy - 0.5f * qdy, y1q = qy + 0.5f * qdy;
  // Faithful to reference: both terms of iw use x0g and x1q.
  float iw = fminf(x0g, x1q) - fmaxf(x0g, x1q) + GIOU_EPS;
  float ih = fminf(y1g, y1q) - fmaxf(y0g, y0q) + GIOU_EPS;
  inter = (iw > 0.0f && ih > 0.0f) ? iw * ih : 0.0f;
  float iwm = fmaxf(x1g, x1q) - fminf(x0g, x0q) + GIOU_EPS;
  float ihm = fmaxf(y1g, y1q) - fminf(y0g, y0q) + GIOU_EPS;
  mbr = iwm * ihm;
}

__global__ void __launch_bounds__(TPB)
gious3d_kernel(const float* __restrict__ g, const float* __restrict__ q,
               float* __restrict__ out, int n) {
  __shared__ float sg[FLOATS_PER_BLOCK];
  __shared__ float sq[FLOATS_PER_BLOCK];

  const int t = threadIdx.x;
  const int row0 = (int)blockIdx.x * ROWS_PER_BLOCK;
  const long long base = (long long)row0 * 7;

  const bool full = (row0 + ROWS_PER_BLOCK) <= n;  // wave-uniform
  if (full) {
    // LDS generic addresses carry the LDS byte offset in the low 32 bits.
    const unsigned sg_off = (unsigned)(size_t)(&sg[0]);
    const unsigned sq_off = (unsigned)(size_t)(&sq[0]);
    const char* gb = (const char*)(g + base);
    const char* qb = (const char*)(q + base);

    // 448 chunks / 256 threads: every thread does chunk t, threads < 192 also
    // do chunk 256+t (EXEC-masked async issue is fine).
    async_copy_b128(sg_off + (unsigned)t * 16u, gb + (size_t)t * 16u);
    async_copy_b128(sq_off + (unsigned)t * 16u, qb + (size_t)t * 16u);
    if (t < (VEC4_PER_BLOCK - TPB)) {
      const int j = TPB + t;
      async_copy_b128(sg_off + (unsigned)j * 16u, gb + (size_t)j * 16u);
      async_copy_b128(sq_off + (unsigned)j * 16u, qb + (size_t)j * 16u);
    }
    asm volatile("s_wait_asynccnt 0" ::: "memory");
  } else {
    // Tail block: bounds-checked scalar staging.
    const long long total = (long long)n * 7;
    for (int j = t; j < FLOATS_PER_BLOCK; j += TPB) {
      long long idx = base + j;
      if (idx < total) {
        sg[j] = g[idx];
        sq[j] = q[idx];
      }
    }
  }
  __syncthreads();

  const int row = row0 + t;
  if (row >= n) return;

  const float* G = &sg[t * 7];
  const float* Q = &sq[t * 7];
  const float g0 = G[0], g1 = G[1], g2 = G[2], g3 = G[3], g4 = G[4], g5 = G[5];
  const float q0 = Q[0], q1 = Q[1], q2 = Q[2], q3 = Q[3], q4 = Q[4], q5 = Q[5];

  // xoz plane: (x,z) with dims (dx,dz)
  float i_xoz, m_xoz;
  plane(g0, g2, g3, g5, q0, q2, q3, q5, i_xoz, m_xoz);
  // xoy plane: (x,y) with dims (dx,dy)  (its mbr is unused; its inter leaks
  // into union_yoz per the reference's bug)
  float i_xoy, m_xoy;
  plane(g0, g1, g3, g4, q0, q1, q3, q4, i_xoy, m_xoy);
  // yoz plane: (y,z) with dims (dy,dz)
  float i_yoz, m_yoz;
  plane(g1, g2, g4, g5, q1, q2, q4, q5, i_yoz, m_yoz);

  const float u_xoz = g3 * g5 + q3 * q5 - i_xoz;
  const float gi_xoz = i_xoz / u_xoz - (m_xoz - u_xoz) / m_xoz;

  // Faithful bug: union_yoz subtracts inter_xoy (not inter_yoz).
  const float u_yoz = g4 * g5 + q4 * q5 - i_xoy;
  const float gi_yoz = i_yoz / u_yoz - (m_yoz - u_yoz) / m_yoz;

  const bool ind = (g3 > 0.0f) && (g4 > 0.0f) && (g5 > 0.0f) &&
                   (q3 > 0.0f) && (q4 > 0.0f) && (q5 > 0.0f);

  // Faithful blend: (gious_xoz + 2*gious_yoz) / 3
  out[row] = ind ? (gi_xoz + 2.0f * gi_yoz) / 3.0f : 0.0f;
}

extern "C" void kernel_launch(void* const* d_in, const int* in_sizes, int n_in,
                              void* d_out, int out_size, void* d_ws,
                              size_t ws_size, hipStream_t stream) {
  (void)n_in;
  (void)d_ws;
  (void)ws_size;
  (void)out_size;
  const float* g = (const float*)d_in[0];
  const float* q = (const float*)d_in[1];
  float* out = (float*)d_out;
  const int n = in_sizes[0] / 7;
  const int blocks = (n + ROWS_PER_BLOCK - 1) / ROWS_PER_BLOCK;
  gious3d_kernel<<<blocks, TPB, 0, stream>>>(g, q, out, n);
}